// TVB_19670950215743
// MI455X (gfx1250) — compile-verified
//
#include <hip/hip_runtime.h>
#include <stdint.h>

// Problem constants (from reference)
#define NN    1024
#define NHH   1024
#define TT    48
#define ROWB  2048            // floats per time slice = N * NSV
#define DT_C  0.1f
#define G_C   0.5f
#define TAU_C 1.0f
#define DEL_C 1.0f
#define J_C   15.0f
#define PI_C  3.14159265358979f

typedef float        v2f   __attribute__((ext_vector_type(2)));
typedef float        v8f   __attribute__((ext_vector_type(8)));
typedef unsigned int u32x4 __attribute__((ext_vector_type(4)));
typedef int          i32x4 __attribute__((ext_vector_type(4)));
typedef int          i32x8 __attribute__((ext_vector_type(8)));

#if defined(__has_builtin)
#  if __has_builtin(__builtin_amdgcn_tensor_load_to_lds) && __has_builtin(__builtin_amdgcn_s_wait_tensorcnt)
#    define USE_TDM 1
#  endif
#  if __has_builtin(__builtin_amdgcn_wmma_f32_16x16x4_f32)
#    define USE_WMMA 1
#  endif
#endif

#ifdef USE_TDM
// 1-D contiguous DMA of n_elems 4-byte elements from global -> LDS via the
// Tensor Data Mover.  D# bit packing per CDNA5 ISA.
// 6-arg builtin form (amdgpu-toolchain / clang-23): groups 2,3,"4" unused->0.
__device__ __forceinline__ void tdm_load_1d(uint32_t lds_byte_off,
                                            const void* gptr,
                                            uint32_t n_elems) {
  uint64_t ga = (uint64_t)(uintptr_t)gptr;
  u32x4 g0;
  g0[0] = 1u;                                   // count=1 (valid), no gather
  g0[1] = lds_byte_off;                         // lds_addr       [63:32]
  g0[2] = (uint32_t)(ga & 0xFFFFFFFFu);         // global_addr lo [95:64]
  g0[3] = (uint32_t)((ga >> 32) & 0x01FFFFFFu)  // global_addr hi [120:96]
        | (2u << 30);                           // type=2 (image) [127:126]
  i32x8 g1;
  g1[0] = (int)(2u << 16);                      // data_size=2 (4B); wg_mask=0
  g1[1] = (int)((n_elems & 0xFFFFu) << 16);     // tensor_dim0 lo16 @bit48
  g1[2] = (int)(((n_elems >> 16) & 0xFFFFu)     // tensor_dim0 hi16
        | (1u << 16));                          // tensor_dim1 = 1  @bit80
  g1[3] = (int)((n_elems & 0xFFFFu) << 16);     // tile_dim0        @bit112
  g1[4] = 1;                                    // tile_dim1=1, tile_dim2=0
  g1[5] = (int)n_elems;                         // tensor_dim0_stride lo @bit160
  g1[6] = 0;                                    // stride hi / dim1_stride lo
  g1[7] = 0;
  i32x4 z4 = {0, 0, 0, 0};
  i32x8 z8 = {0, 0, 0, 0, 0, 0, 0, 0};
  __builtin_amdgcn_tensor_load_to_lds(g0, g1, z4, z4, z8, 0);
}
#endif

__global__ __launch_bounds__(256)
void tvb_step(const float* __restrict__ Wt, const int* __restrict__ lags,
              const int* __restrict__ ixf, const float* __restrict__ region,
              const float* __restrict__ dWt, float* __restrict__ buf,
              float* __restrict__ out, int t) {
  __shared__ float lds_w[NN];
  __shared__ int   lds_l[NN];
  __shared__ int   lds_s[NN];
  __shared__ float p1[256];
  __shared__ float p2[256];

  const int i   = blockIdx.x;       // output row
  const int tid = threadIdx.x;
  const int t0  = NHH + t;          // time index for first coupling

  // ---- Stage descriptor rows (Wt / lags / ix) into LDS ----
#ifdef USE_TDM
  if (tid == 0) {
    tdm_load_1d((uint32_t)(uintptr_t)lds_w, Wt   + (size_t)i * NN, NN);
    tdm_load_1d((uint32_t)(uintptr_t)lds_l, lags + (size_t)i * NN, NN);
    tdm_load_1d((uint32_t)(uintptr_t)lds_s, ixf  + (size_t)i * NN, NN);
    __builtin_amdgcn_s_wait_tensorcnt(0);
  }
  __syncthreads();
#else
#pragma unroll
  for (int k = 0; k < 4; ++k) {
    int j = tid + k * 256;
    lds_w[j] = Wt[(size_t)i * NN + j];
    lds_l[j] = lags[(size_t)i * NN + j];
    lds_s[j] = ixf[(size_t)i * NN + j];
  }
  __syncthreads();
#endif

  // ---- Gather + weighted accumulation (both coupling times in one pass) ----
  float a1 = 0.f, a2 = 0.f;
#pragma unroll
  for (int k = 0; k < 4; ++k) {
    int   j   = tid + k * 256;
    float w   = lds_w[j];
    int   lag = lds_l[j];
    int   src = lds_s[j];
    const float* p = buf + (size_t)(t0 - lag) * ROWB + (src << 1);
    a1 = fmaf(w, p[0],    a1);      // r at time t0-lag
    a2 = fmaf(w, p[ROWB], a2);      // r at time t0+1-lag
  }
  p1[tid] = a1;
  p2[tid] = a2;
  __syncthreads();

  // ---- Reduce 256 partials (wave 0) ----
  if (tid < 32) {                   // wave-uniform branch: EXEC all-ones in wave 0
    float c1, c2;
#ifdef USE_WMMA
    // Matrix-pipe reduction: D += ones(16x4) x B(4x16)  =>  D[m,n] = colsum_n(B).
    // Total sum is permutation-invariant over B's 64 slots, so any lane->slot
    // placement is correct.  4 accumulating WMMAs fold 256 partials -> 16
    // replicated column sums; a 16-lane xor-shuffle finishes.
    const v2f onesA = {1.0f, 1.0f};
    v8f dA = {0.f, 0.f, 0.f, 0.f, 0.f, 0.f, 0.f, 0.f};
    v8f dB = {0.f, 0.f, 0.f, 0.f, 0.f, 0.f, 0.f, 0.f};
#pragma unroll
    for (int c = 0; c < 4; ++c) {
      v2f b1, b2;
      b1[0] = p1[c * 64 + tid];
      b1[1] = p1[c * 64 + 32 + tid];
      b2[0] = p2[c * 64 + tid];
      b2[1] = p2[c * 64 + 32 + tid];
      dA = __builtin_amdgcn_wmma_f32_16x16x4_f32(false, onesA, false, b1,
                                                 (short)0, dA, false, false);
      dB = __builtin_amdgcn_wmma_f32_16x16x4_f32(false, onesA, false, b2,
                                                 (short)0, dB, false, false);
    }
    c1 = dA[0];                     // lane l holds colsum_{l%16}
    c2 = dB[0];
#pragma unroll
    for (int off = 8; off >= 1; off >>= 1) {
      c1 += __shfl_xor(c1, off, 32);
      c2 += __shfl_xor(c2, off, 32);
    }
#else
    c1 = 0.f; c2 = 0.f;
#pragma unroll
    for (int k = 0; k < 8; ++k) {
      c1 += p1[tid + 32 * k];
      c2 += p2[tid + 32 * k];
    }
#pragma unroll
    for (int off = 16; off >= 1; off >>= 1) {
      c1 += __shfl_xor(c1, off, 32);
      c2 += __shfl_xor(c2, off, 32);
    }
#endif

    // ---- Heun update for row i ----
    if (tid == 0) {
      const float eta  = region[i * 2 + 0];
      const float iext = region[i * 2 + 1];
      const float r    = buf[(size_t)t0 * ROWB + (i << 1)];
      const float V    = buf[(size_t)t0 * ROWB + (i << 1) + 1];
      const float n_r  = dWt[(size_t)t * ROWB + (i << 1)];
      const float n_V  = dWt[(size_t)t * ROWB + (i << 1) + 1];
      const float k0   = DEL_C / (PI_C * TAU_C);
      const float itau = 1.0f / TAU_C;

      float dr1 = (k0 + 2.f * r * V) * itau;
      float pr  = PI_C * TAU_C * r;
      float dV1 = (V * V + eta + J_C * TAU_C * r - pr * pr + iext + G_C * c1) * itau;

      float ri = r + DT_C * dr1 + n_r;
      float Vi = V + DT_C * dV1 + n_V;

      float dr2 = (k0 + 2.f * ri * Vi) * itau;
      float pri = PI_C * TAU_C * ri;
      float dV2 = (Vi * Vi + eta + J_C * TAU_C * ri - pri * pri + iext + G_C * c2) * itau;

      float nxr = r + DT_C * 0.5f * (dr1 + dr2) + n_r;
      float nxV = V + DT_C * 0.5f * (dV1 + dV2) + n_V;

      size_t ob = (size_t)(t0 + 1) * ROWB + (i << 1);
      buf[ob]     = nxr;
      buf[ob + 1] = nxV;
      size_t oo = (size_t)t * ROWB + (i << 1);
      out[oo]     = nxr;
      out[oo + 1] = nxV;
    }
  }
}

extern "C" void kernel_launch(void* const* d_in, const int* in_sizes, int n_in,
                              void* d_out, int out_size, void* d_ws, size_t ws_size,
                              hipStream_t stream) {
  (void)in_sizes; (void)n_in; (void)out_size; (void)ws_size;
  const float* Wt     = (const float*)d_in[0];
  const float* buf_in = (const float*)d_in[1];
  const float* dWt    = (const float*)d_in[2];
  const float* region = (const float*)d_in[3];
  const int*   lags   = (const int*)d_in[4];
  const int*   ixf    = (const int*)d_in[5];
  float*       out    = (float*)d_out;

  // Working history buffer lives in scratch (inputs must stay immutable).
  // Slices [0, NH] come from the input; slices NH+1.. are fully rewritten
  // before being read on every call, so no re-initialization is needed.
  float* buf = (float*)d_ws;
  size_t init_bytes = (size_t)(NHH + 1) * ROWB * sizeof(float);
  (void)hipMemcpyAsync(buf, buf_in, init_bytes, hipMemcpyDeviceToDevice, stream);

  for (int t = 0; t < TT; ++t) {
    tvb_step<<<NN, 256, 0, stream>>>(Wt, lags, ixf, region, dWt, buf, out, t);
  }
}